// ContrastiveVolumeLoss_57878979281262
// MI455X (gfx1250) — compile-verified
//
#include <hip/hip_runtime.h>
#include <hip/hip_bf16.h>
#include <math.h>

typedef __attribute__((ext_vector_type(2))) float v2f;
typedef __attribute__((ext_vector_type(8))) float v8f;

#define N_PTS 4096
#define ROWS  8192          // 2*N_PTS
#define CDIM  64
#define HH    256
#define WW    256
#define VV    (HH*WW)
#define INV_T 10.0f
#define EXP_1_OVER_T 22026.465794806718f   // e^10

// ---------------------------------------------------------------------------
// Kernel 1: gather A[row][c] = emb[c*V + ind(row)]   (A is 8192 x 64, f32)
// rows [0,4096) from emb_0 / locations_0, rows [4096,8192) from emb_1 / loc_1
// ---------------------------------------------------------------------------
__global__ void __launch_bounds__(256)
gather_rows_kernel(const float* __restrict__ emb0,
                   const float* __restrict__ emb1,
                   const float* __restrict__ loc0,
                   const float* __restrict__ loc1,
                   float* __restrict__ A) {
  int idx = blockIdx.x * blockDim.x + threadIdx.x;
  if (idx >= ROWS * CDIM) return;
  int row = idx >> 6;   // / CDIM
  int c   = idx & 63;
  const float* emb;
  int ind;
  if (row < N_PTS) {
    float y = floorf(loc0[row * 2 + 0]);
    float x = floorf(loc0[row * 2 + 1]);
    ind = (int)(y * (float)WW + x);
    emb = emb0;
  } else {
    int r = row - N_PTS;
    float y = floorf(loc1[r * 2 + 0]);
    float x = floorf(loc1[r * 2 + 1]);
    ind = (int)(y * (float)WW + x);
    emb = emb1;
  }
  A[idx] = emb[c * VV + ind];
}

// ---------------------------------------------------------------------------
// Kernel 2: per-row sums of exp(A·Aᵀ / T) using V_WMMA_F32_16X16X4_F32.
// One block (8 wave32s) owns one 16-row tile; each wave covers 1024 columns
// (64 column tiles), K-loop of 16 WMMAs covers CDIM=64.
//
// f32 WMMA fragment per lane: row = base + (lane&15), K pair = {2h,2h+1} of
// each K=4 step, h = lane>>4.  Since B = Aᵀ tile, B loads use the same code
// on the column tile's rows.
// ---------------------------------------------------------------------------
__global__ void __launch_bounds__(256)
sim_rowsum_kernel(const float* __restrict__ A, float* __restrict__ rowSum) {
  const int lane = threadIdx.x & 31;
  const int wave = threadIdx.x >> 5;
  const int hi   = lane >> 4;     // which K pair within K=4
  const int lrow = lane & 15;     // row (or col) within tile
  const int rowBase = blockIdx.x * 16;

  // A fragment for this block's row tile: 32 f32 per lane (all K steps)
  float af[32];
  {
    const float* p = A + (size_t)(rowBase + lrow) * CDIM + 2 * hi;
    #pragma unroll
    for (int kk = 0; kk < 16; ++kk) {
      af[2 * kk + 0] = p[4 * kk + 0];
      af[2 * kk + 1] = p[4 * kk + 1];
    }
  }

  float rsum[8];
  #pragma unroll
  for (int r = 0; r < 8; ++r) rsum[r] = 0.0f;

  const int ctBase = wave * 64;           // 64 column tiles per wave
  for (int ct = 0; ct < 64; ++ct) {
    const int colBase = (ctBase + ct) * 16;

    // B fragment = A rows of the column tile (same per-lane pattern)
    float bf[32];
    {
      const float* p = A + (size_t)(colBase + lrow) * CDIM + 2 * hi;
      #pragma unroll
      for (int kk = 0; kk < 16; ++kk) {
        bf[2 * kk + 0] = p[4 * kk + 0];
        bf[2 * kk + 1] = p[4 * kk + 1];
      }
    }

    v8f acc = {};
    #pragma unroll
    for (int kk = 0; kk < 16; ++kk) {
      v2f a; a.x = af[2 * kk]; a.y = af[2 * kk + 1];
      v2f b; b.x = bf[2 * kk]; b.y = bf[2 * kk + 1];
      // 8 args: (neg_a, A, neg_b, B, c_mod, C, reuse_a, reuse_b)
      acc = __builtin_amdgcn_wmma_f32_16x16x4_f32(
          false, a, false, b, (short)0, acc, false, false);
    }

    // acc[r] is element (M = r + 8*hi, N = lrow) of the 16x16 sim tile
    #pragma unroll
    for (int r = 0; r < 8; ++r)
      rsum[r] += __expf(acc[r] * INV_T);
  }

  // Reduce over the 16 lanes that share the same M rows (xor lane bits 0..3)
  #pragma unroll
  for (int r = 0; r < 8; ++r) {
    float s = rsum[r];
    s += __shfl_xor(s, 8, 32);
    s += __shfl_xor(s, 4, 32);
    s += __shfl_xor(s, 2, 32);
    s += __shfl_xor(s, 1, 32);
    rsum[r] = s;
  }

  // Cross-wave reduction in LDS, then write the 16 row sums of this tile
  __shared__ float red[8][16];
  if (lrow == 0) {
    #pragma unroll
    for (int r = 0; r < 8; ++r)
      red[wave][hi * 8 + r] = rsum[r];
  }
  __syncthreads();
  if (threadIdx.x < 16) {
    float t = 0.0f;
    #pragma unroll
    for (int w = 0; w < 8; ++w) t += red[w][threadIdx.x];
    rowSum[rowBase + threadIdx.x] = t;
  }
}

// ---------------------------------------------------------------------------
// Kernel 3: finalize.
// loss = -(1/2n) Σ_u [ 2·dot(A_u, A_{u+n})/T - log(S_u - e^10) - log(S_{u+n} - e^10) ]
// ---------------------------------------------------------------------------
__global__ void __launch_bounds__(256)
loss_kernel(const float* __restrict__ A, const float* __restrict__ rowSum,
            float* __restrict__ out) {
  float acc = 0.0f;
  for (int u = threadIdx.x; u < N_PTS; u += 256) {
    const float* a0 = A + (size_t)u * CDIM;
    const float* a1 = A + (size_t)(u + N_PTS) * CDIM;
    float d = 0.0f;
    #pragma unroll
    for (int c = 0; c < CDIM; ++c) d = fmaf(a0[c], a1[c], d);
    float s0 = rowSum[u]         - EXP_1_OVER_T;
    float s1 = rowSum[u + N_PTS] - EXP_1_OVER_T;
    acc += 2.0f * d * INV_T - __logf(s0) - __logf(s1);
  }
  __shared__ float red[256];
  red[threadIdx.x] = acc;
  __syncthreads();
  for (int s = 128; s > 0; s >>= 1) {
    if (threadIdx.x < s) red[threadIdx.x] += red[threadIdx.x + s];
    __syncthreads();
  }
  if (threadIdx.x == 0) out[0] = -red[0] / (2.0f * (float)N_PTS);
}

// ---------------------------------------------------------------------------
extern "C" void kernel_launch(void* const* d_in, const int* in_sizes, int n_in,
                              void* d_out, int out_size, void* d_ws, size_t ws_size,
                              hipStream_t stream) {
  const float* emb0 = (const float*)d_in[0];
  const float* emb1 = (const float*)d_in[1];
  const float* loc0 = (const float*)d_in[2];
  const float* loc1 = (const float*)d_in[3];
  float* out = (float*)d_out;

  float* A      = (float*)d_ws;          // ROWS*CDIM = 524288 f32 = 2 MB
  float* rowSum = A + (size_t)ROWS * CDIM; // 8192 f32

  gather_rows_kernel<<<(ROWS * CDIM + 255) / 256, 256, 0, stream>>>(
      emb0, emb1, loc0, loc1, A);

  sim_rowsum_kernel<<<ROWS / 16, 256, 0, stream>>>(A, rowSum);

  loss_kernel<<<1, 256, 0, stream>>>(A, rowSum, out);
}